// RoiAlign_44074954391690
// MI455X (gfx1250) — compile-verified
//
#include <hip/hip_runtime.h>
#include <hip/hip_bf16.h>

// RoIAlign (4,256,50,50) + 512 boxes -> (512,256,14,14), f32.
// Bandwidth-bound (~200MB total traffic -> ~9us roofline @ 23.3TB/s).
//
// Per box (one 256-thread block):
//  1. Stage the box's 16x16 feature window (all clamped sample indices
//     provably inside it: box extent <= 199/16 + 2 < 16 cells) into LDS with
//     GLOBAL_LOAD_ASYNC_TO_LDS_B32 (ASYNCcnt path, no VGPR round-trip),
//     128 channels per pass, padded channel stride for bank spread.
//  2. x-interpolation as exact 16(ch) x 16(w) x 16(ix) f32 matmuls via
//     chained V_WMMA_F32_16X16X4_F32 (A from LDS in native 16x4 layout,
//     B = per-lane computed interpolation weights, built once in registers).
//  3. y-interpolation: only 2 nonzero rows per iy -> scalar-weighted tile
//     combine on the dual-issue VALU pipe; y-geometry runs on the scalar
//     float pipe. Validity masks folded into weights (matches reference
//     clamp-coincidence and out-of-range-zero semantics).

typedef __attribute__((ext_vector_type(2))) float v2f;
typedef __attribute__((ext_vector_type(8))) float v8f;

#define N_IMG   4
#define C_CH    256
#define H_FM    50
#define W_FM    50
#define M_BOX   512
#define CROP    14
#define WIN     16
#define CSTRIDE 258   // dwords per channel slab (16*16 + 2 pad: bank spread, keeps b64 8B-aligned)
#define CCHUNK  128   // channels per staging pass; LDS = 128*258*4 = 132096 B <= 320KB/WG

__global__ __launch_bounds__(256, 1) void roi_align_wmma(
    const float* __restrict__ fm, const float* __restrict__ boxes,
    const int* __restrict__ box_idx, float* __restrict__ out)
{
  extern __shared__ float smem[];   // CCHUNK * CSTRIDE floats

  const int m    = blockIdx.x;
  const int t    = threadIdx.x;
  const int lane = t & 31;
  const int wave = t >> 5;
  const int half = (lane >> 4) & 1; // 0: K rows {0,1}, 1: K rows {2,3} (WMMA A/B layout)
  const int ix   = lane & 15;       // output column (N dim); cols 14,15 zero-weight padding

  const int   n   = box_idx[m];
  const float x1f = boxes[m * 4 + 0] * (1.0f / 16.0f);
  const float y1f = boxes[m * 4 + 1] * (1.0f / 16.0f);
  const float x2f = boxes[m * 4 + 2] * (1.0f / 16.0f);
  const float y2f = boxes[m * 4 + 3] * (1.0f / 16.0f);

  int wx0 = (int)floorf(x1f); wx0 = min(max(wx0, 0), W_FM - WIN);
  int wy0 = (int)floorf(y1f); wy0 = min(max(wy0, 0), H_FM - WIN);

  const float dx = (x2f - x1f) * (1.0f / (CROP - 1));
  const float dy = (y2f - y1f) * (1.0f / (CROP - 1));

  // ---- Build B operand (x-interp weights) in registers, per lane ----
  // B is 4x16 (K=w-offset, N=ix). VGPR layout: .x holds K=4*kb+2*half, .y holds K+1.
  v2f B[4];
  {
    float xsv = x1f + (float)ix * dx;
    float xf0 = floorf(xsv);
    float fx  = xsv - xf0;
    int x0c = min(max((int)xf0, 0), W_FM - 1);
    int x1c = min(x0c + 1, W_FM - 1);
    int x0r = x0c - wx0;
    int x1r = x1c - wx0;
    float vx = (ix < CROP && xsv >= 0.0f && xsv <= (float)(W_FM - 1)) ? 1.0f : 0.0f;
    float w0 = (1.0f - fx) * vx;
    float w1 = fx * vx;
#pragma unroll
    for (int kb = 0; kb < 4; ++kb) {
      int we = kb * 4 + 2 * half;
      int wo = we + 1;
      // Sum both terms: reproduces reference clamp-coincidence (x0i==x1i) exactly.
      B[kb].x = (x0r == we ? w0 : 0.0f) + (x1r == we ? w1 : 0.0f);
      B[kb].y = (x0r == wo ? w0 : 0.0f) + (x1r == wo ? w1 : 0.0f);
    }
  }

  const float* fmN = fm + (size_t)n * (C_CH * H_FM * W_FM);
  // Wave-relative LDS byte address of this thread's staging slot: low 32 bits
  // of the generic pointer to LDS are the LDS offset (flat-aperture mapping).
  const unsigned lds0 = (unsigned)(uintptr_t)(smem + t);

  for (int chunk = 0; chunk < 2; ++chunk) {
    const int c0 = chunk * CCHUNK;

    // ---- Stage 128ch x 16 x 16 window into LDS via async DMA ----
    // t -> (r = t>>4, w = t&15): global reads contiguous in w (coalesced),
    // LDS dest addresses lane-consecutive (conflict-free). ASYNCcnt-tracked:
    // no VGPR data round-trip, copy overlaps address generation.
    {
      const int r = t >> 4, w = t & 15;
      const float* src = fmN + ((size_t)c0 * H_FM + (wy0 + r)) * W_FM + (wx0 + w);
#pragma unroll 4
      for (int i = 0; i < CCHUNK; ++i) {
        unsigned    ldsa = lds0 + (unsigned)(i * (CSTRIDE * 4));
        const float* ga  = src + (size_t)i * (H_FM * W_FM);
        asm volatile("global_load_async_to_lds_b32 %0, %1, off"
                     :: "v"(ldsa), "v"(ga) : "memory");
      }
      asm volatile("s_wait_asynccnt 0x0" ::: "memory");
    }
    __syncthreads();

    // ---- Compute: one 16-channel tile per wave ----
    const int ctile = wave * 16;
    const int cA    = ctile + ix;                         // A-matrix row (channel), M = lane&15
    const float* aBase = smem + cA * CSTRIDE + 2 * half;  // +2 dwords for K rows {2,3}

    for (int iy = 0; iy < CROP; ++iy) {
      float ysv = y1f + (float)iy * dy;     // wave-uniform: lowers to scalar-float pipe
      float yf0 = floorf(ysv);
      float fy  = ysv - yf0;
      int y0c = min(max((int)yf0, 0), H_FM - 1);
      int y1c = min(y0c + 1, H_FM - 1);
      int r0  = y0c - wy0;                  // window rows, in [0,15]
      int r1  = y1c - wy0;
      float vy   = (ysv >= 0.0f && ysv <= (float)(H_FM - 1)) ? 1.0f : 0.0f;
      float wy_0 = (1.0f - fy) * vy;
      float wy_1 = fy * vy;

      // g[c, ix] = sum_w fmR[c, r, w] * Bx[w, ix]  -- K=16 via 4 chained WMMAs
      v8f g0 = {};
      v8f g1 = {};
#pragma unroll
      for (int kb = 0; kb < 4; ++kb) {
        v2f a = *(const v2f*)(aBase + r0 * WIN + kb * 4);   // ds_load (2addr_b64 merged)
        g0 = __builtin_amdgcn_wmma_f32_16x16x4_f32(
                 false, a, false, B[kb], (short)0, g0, false, false);
      }
#pragma unroll
      for (int kb = 0; kb < 4; ++kb) {
        v2f a = *(const v2f*)(aBase + r1 * WIN + kb * 4);
        g1 = __builtin_amdgcn_wmma_f32_16x16x4_f32(
                 false, a, false, B[kb], (short)0, g1, false, false);
      }

      // y-interp: 2 nonzero rows -> dual-issue VALU tile combine, clause'd stores.
      // D layout: VGPR j holds M=j (lanes 0-15) / M=j+8 (lanes 16-31), N=lane&15.
      if (ix < CROP) {
        const int cg = c0 + ctile + 8 * half;
        float* op = out + (((size_t)m * C_CH + cg) * CROP + iy) * CROP + ix;
#pragma unroll
        for (int j = 0; j < 8; ++j) {
          float val = g0[j] * wy_0 + g1[j] * wy_1;
          op[(size_t)j * (CROP * CROP)] = val;
        }
      }
    }
    __syncthreads();
  }
}

extern "C" void kernel_launch(void* const* d_in, const int* in_sizes, int n_in,
                              void* d_out, int out_size, void* d_ws, size_t ws_size,
                              hipStream_t stream) {
  (void)in_sizes; (void)n_in; (void)out_size; (void)d_ws; (void)ws_size;
  const float* fm    = (const float*)d_in[0];
  const float* boxes = (const float*)d_in[1];
  const int*   bidx  = (const int*)d_in[2];
  float*       out   = (float*)d_out;

  // 132096 B dynamic LDS per workgroup: within CDNA5's 320KB/WG budget
  // (2 workgroups co-resident per WGP). One block per box.
  const size_t shmem = (size_t)CCHUNK * CSTRIDE * sizeof(float);
  roi_align_wmma<<<dim3(M_BOX), dim3(256), shmem, stream>>>(fm, boxes, bidx, out);
}